// Attention_16612933501279
// MI455X (gfx1250) — compile-verified
//
#include <hip/hip_runtime.h>

typedef float v2f __attribute__((ext_vector_type(2)));
typedef float v8f __attribute__((ext_vector_type(8)));

#define B_  4
#define N_  8
#define C_  256
#define D_  512
#define HW  4096
#define STRIDE 72   // s_tile row stride (floats); 2*72 mod 64 == 16 -> conflict-free B reads

// ---------- precompute kernels (all tiny vs. 12us memory floor) ----------

// qp[b][d] = sum_c Wq[d,c] * q[b,c]        (4*512 outputs, 256 MACs each)
__global__ void k_qp(const float* __restrict__ q, const float* __restrict__ Wq,
                     float* __restrict__ qp) {
    int t = blockIdx.x * blockDim.x + threadIdx.x;   // 2048
    int b = t >> 9, d = t & 511;
    const float* wr = Wq + d * C_;
    const float* qr = q + b * C_;
    float acc = 0.f;
    for (int c = 0; c < C_; ++c) acc += wr[c] * qr[c];
    qp[t] = acc;
}

// gq[b][ci] = g[ci] * sum_d Wkv[d,ci] * qp[b,d]   (K half of Wkv)
__global__ void k_gq(const float* __restrict__ Wkv, const float* __restrict__ g,
                     const float* __restrict__ qp, float* __restrict__ gq) {
    int t = blockIdx.x * blockDim.x + threadIdx.x;   // 1024
    int b = t >> 8, ci = t & 255;
    float acc = 0.f;
    for (int d = 0; d < D_; ++d) acc += Wkv[d * C_ + ci] * qp[b * D_ + d];
    gq[t] = g[ci] * acc;
}

// Fused output matrix A[co][ci] = g[ci] * sum_d Wo[co,d] * Wkv[D+d,ci],
// written pre-swizzled into WMMA 16x4-f32 A-operand layout:
//   element e = ((coTile*64 + k)*32 + lane)*2 + j
//   co = coTile*16 + (lane&15); ci = 4k + ((lane>=16)?2:0) + j
__global__ void k_pa(const float* __restrict__ Wo, const float* __restrict__ Wkv,
                     const float* __restrict__ g, float* __restrict__ pa) {
    int e = blockIdx.x * blockDim.x + threadIdx.x;   // 65536
    int j    = e & 1;
    int lane = (e >> 1) & 31;
    int k    = (e >> 6) & 63;
    int ct   = e >> 12;
    int co = ct * 16 + (lane & 15);
    int ci = 4 * k + ((lane >> 4) << 1) + j;
    const float* wo = Wo + (size_t)co * D_;
    const float* wv = Wkv + (size_t)D_ * C_ + ci;
    float acc = 0.f;
    for (int d = 0; d < D_; ++d) acc += wo[d] * wv[(size_t)d * C_];
    pa[e] = g[ci] * acc;
}

// ---------- main fused kernel ----------
// Grid: 256 WGs of 64 threads (2 waves). WG handles b = wg>>6, 64 pixels
// hwBase..hwBase+63. Thread t = (q4, l): q4 = channel quarter, l -> pixel quad.
__launch_bounds__(64)
__global__ void k_attn(const float* __restrict__ c, const float* __restrict__ gq,
                       const float* __restrict__ pa, const float* __restrict__ bo,
                       float* __restrict__ out) {
    extern __shared__ float lds[];
    float* s_tile = lds;                    // [256][STRIDE]     73728 B
    float* part   = s_tile + 256 * STRIDE;  // [4][64][8][2]     16384 B
    float* w_lds  = part + 4 * 64 * 8 * 2;  // [64][8]            2048 B
    float* gq_lds = w_lds + 64 * 8;         // [256]              1024 B
    float* bo_lds = gq_lds + 256;           // [256]              1024 B

    const int tid    = threadIdx.x;          // 0..63
    const int wg     = blockIdx.x;           // 0..255
    const int b      = wg >> 6;
    const int hwBase = (wg & 63) * 64;
    const int q4     = tid >> 4;             // channel quarter 0..3
    const int l      = tid & 15;             // pixel quad 0..15 (pixels 4l..4l+3)
    const int chBase = q4 * 64;

    for (int i = tid; i < 256; i += 64) {
        gq_lds[i] = gq[b * C_ + i];
        bo_lds[i] = bo[i];
    }
    __syncthreads();

    // ---- pass 1: sumsq and dot(gq) per token, for 4 pixels, over 64 channels ----
    float ss[N_][4], dt[N_][4];
#pragma unroll
    for (int n = 0; n < N_; ++n)
#pragma unroll
        for (int j = 0; j < 4; ++j) { ss[n][j] = 0.f; dt[n][j] = 0.f; }

#pragma unroll 1
    for (int n = 0; n < N_; ++n) {
        const float* cb = c + ((size_t)(b * N_ + n) * C_ + chBase) * HW + hwBase + 4 * l;
#pragma unroll 4
        for (int ch = 0; ch < 64; ++ch) {
            float4 x = *(const float4*)(cb + (size_t)ch * HW);
            float gw = gq_lds[chBase + ch];
            ss[n][0] += x.x * x.x;  dt[n][0] += x.x * gw;
            ss[n][1] += x.y * x.y;  dt[n][1] += x.y * gw;
            ss[n][2] += x.z * x.z;  dt[n][2] += x.z * gw;
            ss[n][3] += x.w * x.w;  dt[n][3] += x.w * gw;
        }
    }
#pragma unroll
    for (int j = 0; j < 4; ++j)
#pragma unroll
        for (int n = 0; n < N_; ++n) {
            int idx = ((q4 * 64 + 4 * l + j) * N_ + n) * 2;
            part[idx]     = ss[n][j];
            part[idx + 1] = dt[n][j];
        }
    __syncthreads();

    // ---- per-pixel softmax (thread tid owns pixel tid) ----
    {
        const int pix = tid;
        float dots[N_], rms[N_];
#pragma unroll
        for (int n = 0; n < N_; ++n) {
            float s = 0.f, d = 0.f;
#pragma unroll
            for (int qq = 0; qq < 4; ++qq) {
                int idx = ((qq * 64 + pix) * N_ + n) * 2;
                s += part[idx]; d += part[idx + 1];
            }
            rms[n]  = rsqrtf(s * (1.0f / 256.0f) + 1e-6f);
            dots[n] = d * rms[n] * 0.04419417382415922f;   // D^-0.5, D=512
        }
        float m = dots[0];
#pragma unroll
        for (int n = 1; n < N_; ++n) m = fmaxf(m, dots[n]);
        float e[N_], sum = 0.f;
#pragma unroll
        for (int n = 0; n < N_; ++n) { e[n] = __expf(dots[n] - m); sum += e[n]; }
        float inv = 1.0f / sum;
#pragma unroll
        for (int n = 0; n < N_; ++n) w_lds[pix * N_ + n] = e[n] * inv * rms[n];
    }
    __syncthreads();

    // ---- pass 2: s[ci] = sum_n w_n * cp[n,ci] -> s_tile[ci][pix] (L2-resident re-read) ----
    float wv[N_][4];
#pragma unroll
    for (int j = 0; j < 4; ++j)
#pragma unroll
        for (int n = 0; n < N_; ++n) wv[n][j] = w_lds[(4 * l + j) * N_ + n];

#pragma unroll 2
    for (int ch = 0; ch < 64; ++ch) {
        const int cg = chBase + ch;
        float4 s4 = make_float4(0.f, 0.f, 0.f, 0.f);
#pragma unroll
        for (int n = 0; n < N_; ++n) {
            const float* cb = c + ((size_t)(b * N_ + n) * C_ + cg) * HW + hwBase + 4 * l;
            float4 x = *(const float4*)cb;
            s4.x += wv[n][0] * x.x;
            s4.y += wv[n][1] * x.y;
            s4.z += wv[n][2] * x.z;
            s4.w += wv[n][3] * x.w;
        }
        *(float4*)(&s_tile[cg * STRIDE + 4 * l]) = s4;
    }
    __syncthreads();

    // ---- fused output projection via V_WMMA_F32_16X16X4_F32 ----
    // D[M=co, N=pix] = sum_ci A[co,ci] * s_tile[ci][pix]
    // Two co-tiles per iteration: each B operand (LDS) feeds 2 WMMAs.
    const int lane  = tid & 31;
    const int wave  = tid >> 5;
    const int laneM = lane & 15;
    const int hiK   = (lane >> 4) << 1;           // 0 or 2

#pragma unroll 1
    for (int i = 0; i < 4; ++i) {                 // 4 co-tile pairs per wave
        const int ct0 = i * 4 + wave * 2;         // wave0: 0,4,8,12  wave1: 2,6,10,14
        v8f acc[2][4] = {};
        const v2f* paP0 = (const v2f*)pa + (size_t)(ct0 * 64) * 32 + lane;
        const v2f* paP1 = paP0 + 64 * 32;         // ct0+1
#pragma unroll 4
        for (int k = 0; k < 64; ++k) {
            v2f a0 = paP0[(size_t)k * 32];        // coalesced b64, pre-swizzled A layout
            v2f a1 = paP1[(size_t)k * 32];
            const int row = 4 * k + hiK;
#pragma unroll
            for (int pt = 0; pt < 4; ++pt) {
                v2f bv;
                bv.x = s_tile[row * STRIDE + pt * 16 + laneM];
                bv.y = s_tile[(row + 1) * STRIDE + pt * 16 + laneM];
                acc[0][pt] = __builtin_amdgcn_wmma_f32_16x16x4_f32(
                    false, a0, false, bv, (short)0, acc[0][pt], false, false);
                acc[1][pt] = __builtin_amdgcn_wmma_f32_16x16x4_f32(
                    false, a1, false, bv, (short)0, acc[1][pt], false, false);
            }
        }
#pragma unroll
        for (int u = 0; u < 2; ++u) {
            const int coB = (ct0 + u) * 16 + ((lane >> 4) << 3);   // +8 for hi half-wave
#pragma unroll
            for (int pt = 0; pt < 4; ++pt) {
                float* op = out + ((size_t)(b * C_ + coB)) * HW + hwBase + pt * 16 + laneM;
#pragma unroll
                for (int r = 0; r < 8; ++r) {
                    op[(size_t)r * HW] = acc[u][pt][r] + bo_lds[coB + r];
                }
            }
        }
    }
}

extern "C" void kernel_launch(void* const* d_in, const int* in_sizes, int n_in,
                              void* d_out, int out_size, void* d_ws, size_t ws_size,
                              hipStream_t stream) {
    (void)in_sizes; (void)n_in; (void)out_size; (void)ws_size;
    const float* q   = (const float*)d_in[0];
    const float* c   = (const float*)d_in[1];
    const float* g   = (const float*)d_in[2];
    const float* Wq  = (const float*)d_in[3];
    const float* Wkv = (const float*)d_in[4];
    const float* Wo  = (const float*)d_in[5];
    const float* bo  = (const float*)d_in[6];
    float* out = (float*)d_out;

    float* ws = (float*)d_ws;
    float* qp = ws;            // 2048 floats
    float* gq = ws + 2048;     // 1024 floats
    float* pa = ws + 3072;     // 65536 floats  (total 274 KB of ws)

    k_qp<<<8,   256, 0, stream>>>(q, Wq, qp);
    k_gq<<<4,   256, 0, stream>>>(Wkv, g, qp, gq);
    k_pa<<<256, 256, 0, stream>>>(Wo, Wkv, g, pa);

    const size_t ldsBytes = (256 * STRIDE + 4 * 64 * 8 * 2 + 64 * 8 + 256 + 256) * sizeof(float);
    k_attn<<<256, 64, ldsBytes, stream>>>(c, gq, pa, bo, out);
}